// MPS_31456340476203
// MI455X (gfx1250) — compile-verified
//
#include <hip/hip_runtime.h>
#include <hip/hip_bf16.h>

typedef float v2f __attribute__((ext_vector_type(2)));
typedef float v8f __attribute__((ext_vector_type(8)));

#define BOND 64
#define MSZ  (BOND * BOND)   // 4096 floats per 64x64 matrix
#define PHYS 4

// ---------------------------------------------------------------------------
// Kernel 1: M_i[l][r] = sum_p A_mid[i][l][p][r]   (fold the physical index)
// A_mid layout: (nmid, 64, 4, 64) row-major. Output: nmid row-major 64x64.
// Coalesced across r (fastest dim).
// ---------------------------------------------------------------------------
__global__ __launch_bounds__(256) void mps_reduce_p(const float* __restrict__ A_mid,
                                                    float* __restrict__ M,
                                                    int total) {
    int idx = blockIdx.x * blockDim.x + threadIdx.x;
    if (idx >= total) return;
    int site = idx >> 12;          // / 4096
    int rem  = idx & 4095;
    int l    = rem >> 6;
    int r    = rem & 63;
    const float* a = A_mid + (size_t)site * 16384 + l * 256 + r;  // stride over p is 64
    M[idx] = (a[0] + a[64]) + (a[128] + a[192]);
}

// ---------------------------------------------------------------------------
// Kernel 2: pairwise 64x64x64 f32 GEMM via V_WMMA_F32_16X16X4_F32.
// One workgroup (128 threads = 4 wave32s) per pair. Wave w computes the
// 16-row stripe D[16w:16w+16, 0:64] as 4 accumulators of 16x16, iterating
// K in steps of 4 (16 WMMA k-steps, 256 WMMA issues per workgroup).
//
// Fragment layouts per CDNA5 ISA 7.12.2:
//   A 16x4 f32 : lanes 0-15 -> K={0,1}, lanes 16-31 -> K={2,3} (v2f)
//   B 4x16 f32 : lanes 0-15 -> rows K={0,1}, lanes 16-31 -> rows K={2,3}
//   C/D 16x16  : VGPR j holds row m=j (lanes 0-15) / m=j+8 (lanes 16-31)
// Odd trailing matrix passes through unchanged (preserves chain order).
// ---------------------------------------------------------------------------
__global__ __launch_bounds__(128) void mps_pairmul(const float* __restrict__ src,
                                                   float* __restrict__ dst,
                                                   int count) {
    int pair = blockIdx.x;
    const float* L = src + (size_t)(2 * pair) * MSZ;
    float* D = dst + (size_t)pair * MSZ;

    if (2 * pair + 1 >= count) {  // uniform across the block: copy-through
        for (int k = threadIdx.x; k < MSZ; k += 128) D[k] = L[k];
        return;
    }
    const float* R = src + (size_t)(2 * pair + 1) * MSZ;

    const int lane = threadIdx.x & 31;
    const int wv   = threadIdx.x >> 5;
    const int half = lane >> 4;        // 0 for lanes 0-15, 1 for 16-31
    const int koff = half << 1;        // K offset 0 or 2 within a k-step
    const int m16  = lane & 15;
    const int mrow = (wv << 4) + m16;  // this lane's A-matrix row

    v8f acc0 = {}, acc1 = {}, acc2 = {}, acc3 = {};

    for (int k0 = 0; k0 < BOND; k0 += 4) {
        v2f a;
        a.x = L[mrow * BOND + k0 + koff];
        a.y = L[mrow * BOND + k0 + koff + 1];

        const float* r0 = R + (k0 + koff) * BOND + m16;  // B row k0+koff
        const float* r1 = r0 + BOND;                     // B row k0+koff+1
        v2f b0, b1, b2, b3;
        b0.x = r0[0];  b0.y = r1[0];
        b1.x = r0[16]; b1.y = r1[16];
        b2.x = r0[32]; b2.y = r1[32];
        b3.x = r0[48]; b3.y = r1[48];

        acc0 = __builtin_amdgcn_wmma_f32_16x16x4_f32(false, a, false, b0, (short)0, acc0, false, false);
        acc1 = __builtin_amdgcn_wmma_f32_16x16x4_f32(false, a, false, b1, (short)0, acc1, false, false);
        acc2 = __builtin_amdgcn_wmma_f32_16x16x4_f32(false, a, false, b2, (short)0, acc2, false, false);
        acc3 = __builtin_amdgcn_wmma_f32_16x16x4_f32(false, a, false, b3, (short)0, acc3, false, false);
    }

    const int rbase = (wv << 4) + (half << 3);  // D row base for this lane
    #pragma unroll
    for (int jj = 0; jj < 8; ++jj) {
        float* drow = D + (rbase + jj) * BOND + m16;
        drow[0]  = acc0[jj];
        drow[16] = acc1[jj];
        drow[32] = acc2[jj];
        drow[48] = acc3[jj];
    }
}

// ---------------------------------------------------------------------------
// Kernel 3: c = m_first @ Mtot @ m_last   (one block, 64 threads)
//   m_first[r] = sum_p A_first[0][p][r]   (A_first: (1,4,64))
//   m_last[l]  = sum_p A_last[l][p][0]    (A_last : (64,4,1))
// ---------------------------------------------------------------------------
__global__ __launch_bounds__(64) void mps_combine(const float* __restrict__ A_first,
                                                  const float* __restrict__ A_last,
                                                  const float* __restrict__ Mtot,
                                                  float* __restrict__ c_out) {
    __shared__ float f[BOND];
    __shared__ float red[BOND];
    const int t = threadIdx.x;  // 0..63

    f[t] = ((A_first[t] + A_first[64 + t]) + (A_first[128 + t] + A_first[192 + t]));
    __syncthreads();

    float g = (A_last[t * 4 + 0] + A_last[t * 4 + 1]) + (A_last[t * 4 + 2] + A_last[t * 4 + 3]);

    float acc = 0.0f;
    #pragma unroll 8
    for (int l = 0; l < BOND; ++l) acc += f[l] * Mtot[l * BOND + t];
    red[t] = acc * g;
    __syncthreads();

    for (int s = 32; s > 0; s >>= 1) {
        if (t < s) red[t] += red[t + s];
        __syncthreads();
    }
    if (t == 0) c_out[0] = red[0];
}

// ---------------------------------------------------------------------------
// Kernel 4: out[b] = c * prod_{i=0}^{127} (x[b][i][0]+x[b][i][1]+x[b][i][2]+x[b][i][3])
// One thread per batch element: streams its contiguous 2 KB with b128 loads
// (every 128B line fetched exactly once); 4 independent product accumulators
// break the serial multiply chain; prefetch ahead by 512 B.
// ---------------------------------------------------------------------------
__global__ __launch_bounds__(256) void mps_batch(const float4* __restrict__ x,
                                                 const float* __restrict__ c_ptr,
                                                 float* __restrict__ out) {
    const int b = blockIdx.x * 256 + threadIdx.x;
    const float4* p = x + (size_t)b * 128;

    float p0 = 1.0f, p1 = 1.0f, p2 = 1.0f, p3 = 1.0f;
    #pragma unroll 4
    for (int i = 0; i < 128; i += 4) {
        __builtin_prefetch(p + i + 32, 0, 0);  // -> global_prefetch_b8, ~512B ahead
        float4 v0 = p[i + 0];
        float4 v1 = p[i + 1];
        float4 v2 = p[i + 2];
        float4 v3 = p[i + 3];
        p0 *= (v0.x + v0.y) + (v0.z + v0.w);
        p1 *= (v1.x + v1.y) + (v1.z + v1.w);
        p2 *= (v2.x + v2.y) + (v2.z + v2.w);
        p3 *= (v3.x + v3.y) + (v3.z + v3.w);
    }
    out[b] = c_ptr[0] * ((p0 * p1) * (p2 * p3));
}

// ---------------------------------------------------------------------------
extern "C" void kernel_launch(void* const* d_in, const int* in_sizes, int n_in,
                              void* d_out, int out_size, void* d_ws, size_t ws_size,
                              hipStream_t stream) {
    const float* x       = (const float*)d_in[0];  // (65536, 128, 4)
    const float* A_first = (const float*)d_in[1];  // (1, 4, 64)
    const float* A_mid   = (const float*)d_in[2];  // (126, 64, 4, 64)
    const float* A_last  = (const float*)d_in[3];  // (64, 4, 1)
    float* out = (float*)d_out;                    // (65536, 1)

    const int nmid = in_sizes[2] / (BOND * PHYS * BOND);  // 126

    // Workspace layout (floats): buf0[nmid*4096] | buf1[ceil(nmid/2)*4096] | c[1]
    float* buf0  = (float*)d_ws;
    float* buf1  = buf0 + (size_t)nmid * MSZ;
    float* c_ptr = buf1 + (size_t)((nmid + 1) / 2) * MSZ;

    // 1) fold physical index out of every mid-site tensor
    const int total = nmid * MSZ;
    mps_reduce_p<<<(total + 255) / 256, 256, 0, stream>>>(A_mid, buf0, total);

    // 2) tree-reduce the ordered matrix chain: 126 -> 63 -> 32 -> ... -> 1
    int count = nmid;
    float* cur = buf0;
    float* nxt = buf1;
    while (count > 1) {
        const int pairs = (count + 1) >> 1;
        mps_pairmul<<<pairs, 128, 0, stream>>>(cur, nxt, count);
        float* tmp = cur; cur = nxt; nxt = tmp;
        count = pairs;
    }

    // 3) boundary vectors -> scalar c
    mps_combine<<<1, 64, 0, stream>>>(A_first, A_last, cur, c_ptr);

    // 4) bandwidth-bound batch sweep: out[b] = c * prod of per-site sums
    mps_batch<<<65536 / 256, 256, 0, stream>>>((const float4*)x, c_ptr, out);
}